// SignedGraphEncoder_2224793060076
// MI455X (gfx1250) — compile-verified
//
#include <hip/hip_runtime.h>

// ---------------- problem constants (from reference) ----------------
#define NUM_USERS 200000
#define NUM_ITEMS 100000
#define NNODES    (NUM_USERS + NUM_ITEMS)   // 300000
#define EMBED     64
#define ALPHA_K   (1.0f / 3.0f)             // 1/(NUM_LAYERS+1), NUM_LAYERS=2

#define BLOCK      256
#define TILE_EDGES 1024                      // edges staged in LDS per block

// ---------------- CDNA5 async global->LDS helpers ----------------
typedef __attribute__((ext_vector_type(4))) int v4i;
typedef __attribute__((address_space(1))) v4i gv4i;   // global
typedef __attribute__((address_space(3))) v4i lv4i;   // LDS

__device__ __forceinline__ void async_copy16(const void* gsrc, void* ldst) {
  // one VMEM instruction per wave: each lane moves 16B global -> LDS (ASYNCcnt)
  __builtin_amdgcn_global_load_async_to_lds_b128(
      (gv4i*)gsrc, (lv4i*)ldst, /*imm offset*/ 0, /*cpol*/ 0);
}

__device__ __forceinline__ void wait_async0() {
  __builtin_amdgcn_s_wait_asynccnt(0);
}

__device__ __forceinline__ void atomic_add_f32(float* p, float v) {
  // native global_atomic_add_f32, non-returning
  unsafeAtomicAdd(p, v);
}

// ---------------- kernels ----------------

// deg[col] += 1 for every edge
__global__ void deg_kernel(const int* __restrict__ cols, float* __restrict__ deg, int E) {
  int e = blockIdx.x * blockDim.x + threadIdx.x;
  if (e < E) atomic_add_f32(&deg[cols[e]], 1.0f);
}

// deg -> dinv in place
__global__ void dinv_kernel(float* __restrict__ deg, int n) {
  int i = blockIdx.x * blockDim.x + threadIdx.x;
  if (i < n) {
    float d = deg[i];
    deg[i] = (d > 0.0f) ? rsqrtf(d) : 0.0f;
  }
}

// norm[e] = dinv[row[e]] * dinv[col[e]]  (layer-invariant, computed once per sign)
__global__ void norm_kernel(const int* __restrict__ rows, const int* __restrict__ cols,
                            const float* __restrict__ dinv, float* __restrict__ norm, int E) {
  int e = blockIdx.x * blockDim.x + threadIdx.x;
  if (e < E) norm[e] = dinv[rows[e]] * dinv[cols[e]];
}

// buf = concat(user, item); out = ALPHA * buf; z = 0   (float4 granularity)
__global__ void concat_init_kernel(const float4* __restrict__ user,
                                   const float4* __restrict__ item,
                                   float4* __restrict__ buf,
                                   float4* __restrict__ out,
                                   float4* __restrict__ z,
                                   int n4u, int n4) {
  int i = blockIdx.x * blockDim.x + threadIdx.x;
  if (i < n4) {
    float4 v = (i < n4u) ? user[i] : item[i - n4u];
    buf[i] = v;
    float4 s;
    s.x = v.x * ALPHA_K; s.y = v.y * ALPHA_K;
    s.z = v.z * ALPHA_K; s.w = v.w * ALPHA_K;
    out[i] = s;
    float4 zero = {0.f, 0.f, 0.f, 0.f};
    z[i] = zero;
  }
}

// out += ALPHA * y; z = 0   (float4 granularity; zeroes next conv's output buffer)
__global__ void axpy_zero_kernel(float4* __restrict__ out, const float4* __restrict__ y,
                                 float4* __restrict__ z, int n4) {
  int i = blockIdx.x * blockDim.x + threadIdx.x;
  if (i < n4) {
    float4 o = out[i];
    float4 v = y[i];
    o.x += v.x * ALPHA_K; o.y += v.y * ALPHA_K;
    o.z += v.z * ALPHA_K; o.w += v.w * ALPHA_K;
    out[i] = o;
    float4 zero = {0.f, 0.f, 0.f, 0.f};
    z[i] = zero;
  }
}

// out += ALPHA * y   (float4 granularity; final layer)
__global__ void axpy_kernel(float4* __restrict__ out, const float4* __restrict__ y, int n4) {
  int i = blockIdx.x * blockDim.x + threadIdx.x;
  if (i < n4) {
    float4 o = out[i];
    float4 v = y[i];
    o.x += v.x * ALPHA_K; o.y += v.y * ALPHA_K;
    o.z += v.z * ALPHA_K; o.w += v.w * ALPHA_K;
    out[i] = o;
  }
}

// y[col] += norm[e] * x[row]  for each staged edge.
// 16 lanes per edge, float4 per lane (64 floats). Edge indices + norms staged
// to LDS via async b128 copies. Requires E % 4 == 0 (true: 1,000,000 / 500,000).
__global__ void conv_kernel(const float4* __restrict__ x4,
                            float* __restrict__ y,
                            const int* __restrict__ rows,
                            const int* __restrict__ cols,
                            const float* __restrict__ norm,
                            int E) {
  __shared__ int   lds_rows[TILE_EDGES];
  __shared__ int   lds_cols[TILE_EDGES];
  __shared__ float lds_norm[TILE_EDGES];

  const int t    = threadIdx.x;
  const int base = blockIdx.x * TILE_EDGES;

  // ---- stage tile of edge data into LDS (async, ASYNCcnt-tracked) ----
  {
    int s = base + t * 4;            // 4 elements (16B) per lane
    int smax = E - 4;
    if (s > smax) s = smax;          // clamp (E multiple of 4); dup slots never read
    async_copy16(rows + s, &lds_rows[t * 4]);
    async_copy16(cols + s, &lds_cols[t * 4]);
    async_copy16(norm + s, &lds_norm[t * 4]);
    wait_async0();
  }
  __syncthreads();

  const int group = t >> 4;          // 16 groups of 16 lanes per block
  const int sub   = t & 15;          // float4 slot within the 64-float row

  #pragma unroll 4
  for (int pass = 0; pass < TILE_EDGES / 16; ++pass) {
    int el = pass * 16 + group;
    int e  = base + el;
    if (e < E) {
      int   r = lds_rows[el];
      int   c = lds_cols[el];
      float w = lds_norm[el];
      float4 v = x4[(size_t)r * (EMBED / 4) + sub];
      float* yp = y + (size_t)c * EMBED + sub * 4;
      atomic_add_f32(yp + 0, w * v.x);
      atomic_add_f32(yp + 1, w * v.y);
      atomic_add_f32(yp + 2, w * v.z);
      atomic_add_f32(yp + 3, w * v.w);
    }
  }
}

// ---------------- host orchestration ----------------
extern "C" void kernel_launch(void* const* d_in, const int* in_sizes, int n_in,
                              void* d_out, int out_size, void* d_ws, size_t ws_size,
                              hipStream_t stream) {
  const float* user_pos = (const float*)d_in[0];
  const float* item_pos = (const float*)d_in[1];
  const float* user_neg = (const float*)d_in[2];
  const float* item_neg = (const float*)d_in[3];
  const int*   pos_ei   = (const int*)d_in[4];
  const int*   neg_ei   = (const int*)d_in[5];
  const int E_pos = in_sizes[4] / 2;
  const int E_neg = in_sizes[5] / 2;

  float* out = (float*)d_out;

  const size_t embN = (size_t)NNODES * EMBED;         // 19.2M floats
  float* bufA = (float*)d_ws;
  float* bufB = bufA + embN;
  float* deg  = bufB + embN;                          // NNODES floats
  float* norm = deg + NNODES;                         // up to max(E) floats

  const int n4  = (int)(embN / 4);
  const int n4u = NUM_USERS * EMBED / 4;
  const int g4  = (n4 + BLOCK - 1) / BLOCK;

  struct Sign { const float* u; const float* it; const int* ei; int E; float* osec; };
  Sign signs[2] = {
    { user_pos, item_pos, pos_ei, E_pos, out          },
    { user_neg, item_neg, neg_ei, E_neg, out + embN   },
  };

  for (int s = 0; s < 2; ++s) {
    const int* rows = signs[s].ei;
    const int* cols = signs[s].ei + signs[s].E;
    const int  E    = signs[s].E;
    float*     osec = signs[s].osec;
    const int  gE   = (E + BLOCK - 1) / BLOCK;

    // degrees -> dinv (stored in deg) -> per-edge norms
    (void)hipMemsetAsync(deg, 0, NNODES * sizeof(float), stream);
    deg_kernel<<<gE, BLOCK, 0, stream>>>(cols, deg, E);
    dinv_kernel<<<(NNODES + BLOCK - 1) / BLOCK, BLOCK, 0, stream>>>(deg, NNODES);
    norm_kernel<<<gE, BLOCK, 0, stream>>>(rows, cols, deg, norm, E);

    // x0 = concat(user, item); out = alpha * x0; bufB = 0
    concat_init_kernel<<<g4, BLOCK, 0, stream>>>(
        (const float4*)signs[s].u, (const float4*)signs[s].it,
        (float4*)bufA, (float4*)osec, (float4*)bufB, n4u, n4);

    const int cblocks = (E + TILE_EDGES - 1) / TILE_EDGES;

    // layer 1: x1 = conv(x0); out += alpha * x1; bufA = 0
    conv_kernel<<<cblocks, BLOCK, 0, stream>>>((const float4*)bufA, bufB, rows, cols, norm, E);
    axpy_zero_kernel<<<g4, BLOCK, 0, stream>>>((float4*)osec, (const float4*)bufB, (float4*)bufA, n4);

    // layer 2: x2 = conv(x1); out += alpha * x2
    conv_kernel<<<cblocks, BLOCK, 0, stream>>>((const float4*)bufB, bufA, rows, cols, norm, E);
    axpy_kernel<<<g4, BLOCK, 0, stream>>>((float4*)osec, (const float4*)bufA, n4);
  }
}